// ROIHead_63101659512967
// MI455X (gfx1250) — compile-verified
//
#include <hip/hip_runtime.h>
#include <hip/hip_bf16.h>
#include <math.h>

typedef float v2f __attribute__((ext_vector_type(2)));
typedef float v8f __attribute__((ext_vector_type(8)));

#define RPOOL 7
#define RSUB 2
#define RS 14            // RPOOL*RSUB
#define RC 256
#define RD 12544         // RC*RPOOL*RPOOL
#define NROIS 1000

// ---------------------------------------------------------------------------
// Kernel 1: pyramid ROI-align.  One block per ROI, 256 threads.
// Writes x[roi][c*49 + py*7 + px] (row-major, matches pooled.reshape(N,-1)).
// ---------------------------------------------------------------------------
__global__ void roihead_roialign_kernel(const float* __restrict__ f2,
                                        const float* __restrict__ f3,
                                        const float* __restrict__ f4,
                                        const float* __restrict__ f5,
                                        const float* __restrict__ rois,
                                        const int*   __restrict__ img_size,
                                        float* __restrict__ xout)
{
    const int roi = blockIdx.x;

    __shared__ int   s_iu[RS], s_idn[RS], s_il[RS], s_ir[RS];
    __shared__ float s_ly[RS], s_lx[RS];

    // Every thread redundantly computes the (uniform) ROI/level parameters.
    const float r0 = rois[roi * 4 + 0];
    const float r1 = rois[roi * 4 + 1];
    const float r2 = rois[roi * 4 + 2];
    const float r3 = rois[roi * 4 + 3];

    const float hh = r2 - r0 + 1.0f;
    const float ww = r3 - r1 + 1.0f;
    float lvlf = floorf(logf(sqrtf(hh * ww) / 224.0f) / 0.693147f + 4.0f);
    lvlf = fminf(fmaxf(lvlf, 2.0f), 5.0f);
    const int lvl = (int)lvlf;

    int H;
    const float* f;
    if (lvl == 2)      { H = 256; f = f2; }
    else if (lvl == 3) { H = 128; f = f3; }
    else if (lvl == 4) { H = 64;  f = f4; }
    else               { H = 32;  f = f5; }

    const float imh = (float)img_size[0] - 1.0f;
    const float imw = (float)img_size[1] - 1.0f;
    const float fh  = (float)H - 1.0f;   // all levels square: W == H

    const float y1 = r0 * fh / imh;
    const float x1 = r1 * fh / imw;
    const float y2 = r2 * fh / imh;
    const float x2 = r3 * fh / imw;
    const float hstep = (y2 - y1) / (float)RS;
    const float wstep = (x2 - x1) / (float)RS;

    if (threadIdx.x < RS) {
        const int g = threadIdx.x;
        const float cy = ((float)g + 0.5f) * hstep + y1;
        const float cx = ((float)g + 0.5f) * wstep + x1;
        float fy = floorf(cy), fx = floorf(cx);
        int iu  = (int)fy;
        int idn = (int)ceilf(cy);
        int il  = (int)fx;
        int ir  = (int)ceilf(cx);
        // safety clamp (reference coords are always in range)
        iu  = max(0, min(iu,  H - 1));
        idn = max(0, min(idn, H - 1));
        il  = max(0, min(il,  H - 1));
        ir  = max(0, min(ir,  H - 1));
        s_iu[g] = iu; s_idn[g] = idn; s_ly[g] = cy - fy;
        s_il[g] = il; s_ir[g]  = ir;  s_lx[g] = cx - fx;
    }
    __syncthreads();

    const int HW = H * H;
    float* orow = xout + (size_t)roi * RD;

    for (int i = threadIdx.x; i < RD; i += blockDim.x) {
        const int c  = i / 49;
        const int p  = i - c * 49;
        const int py = p / 7;
        const int px = p - py * 7;
        const float* fc = f + (size_t)c * HW;

        float m = -INFINITY;
#pragma unroll
        for (int sy = 0; sy < RSUB; ++sy) {
#pragma unroll
            for (int sx = 0; sx < RSUB; ++sx) {
                const int gy = py * RSUB + sy;
                const int gx = px * RSUB + sx;
                const int iu = s_iu[gy], idn = s_idn[gy];
                const int il = s_il[gx], ir  = s_ir[gx];
                const float ly = s_ly[gy], lx = s_lx[gx];
                const float v =
                    fc[iu  * H + il] * (1.0f - ly) * (1.0f - lx) +
                    fc[idn * H + il] * ly          * (1.0f - lx) +
                    fc[iu  * H + ir] * (1.0f - ly) * lx +
                    fc[idn * H + ir] * ly          * lx;
                m = fmaxf(m, v);
            }
        }
        orow[i] = m;
    }
}

// ---------------------------------------------------------------------------
// Kernel 2: fp32 WMMA GEMM  C = relu?(A(MxK) * B(KxN) + bias)  with M = NROIS.
// N, K, RELU are compile-time -> all B strides become immediate offsets and
// the hot loop reduces to 2 pointer bumps + 9 loads + 4 WMMAs.
// Wave computes 16(M) x 64(N): 4 independent V_WMMA_F32_16X16X4_F32 accums;
// block = 4 waves -> 64x64 tile (waves share B addresses -> WGP$ hits).
// A frag (16x4): lanes 0-15 hold K={0,1}, lanes 16-31 K={2,3} (ISA 7.12.2).
// B frag (4x16): mirrored.  C/D: VGPR i -> row i (lanes 0-15) / i+8 (16-31).
// Edges handled by index clamping (no per-lane predication in the hot loop);
// only the stores are guarded.  K loop software-pipelined.
// ---------------------------------------------------------------------------
#define WMMA_F32(acc, a, b) \
    __builtin_amdgcn_wmma_f32_16x16x4_f32(false, (a), false, (b), (short)0, (acc), false, false)

template <int N, int K, int RELU, bool EDGE>
__device__ __forceinline__ void gemm_wmma_body(const float* __restrict__ A,
                                               const float* __restrict__ B,
                                               const float* __restrict__ bias,
                                               float* __restrict__ Cout,
                                               int m0, int n0)
{
    const int lane = threadIdx.x;          // 0..31
    const int half = lane >> 4;            // 0: K {0,1} ; 1: K {2,3}
    const int r    = lane & 15;

    int mrow = m0 + r;
    if (mrow > NROIS - 1) mrow = NROIS - 1;   // clamp loads; stores are guarded
    const float* ap = A + (size_t)mrow * K + 2 * half;

    v8f acc0 = {}, acc1 = {}, acc2 = {}, acc3 = {};

    if (!EDGE) {
        // Single running pointer; all other B accesses via constant offsets.
        const float* p = B + (size_t)(2 * half) * N + (n0 + r);

        v2f a_c = *(const v2f*)ap;  ap += 4;
        v2f b0_c, b1_c, b2_c, b3_c;
        b0_c.x = p[0];  b0_c.y = p[N];
        b1_c.x = p[16]; b1_c.y = p[N + 16];
        b2_c.x = p[32]; b2_c.y = p[N + 32];
        b3_c.x = p[48]; b3_c.y = p[N + 48];
        p += 4 * N;

        for (int k = 0; k + 4 < K; k += 4) {
            const v2f a_n = *(const v2f*)ap;  ap += 4;
            v2f b0_n, b1_n, b2_n, b3_n;
            b0_n.x = p[0];  b0_n.y = p[N];
            b1_n.x = p[16]; b1_n.y = p[N + 16];
            b2_n.x = p[32]; b2_n.y = p[N + 32];
            b3_n.x = p[48]; b3_n.y = p[N + 48];
            p += 4 * N;

            acc0 = WMMA_F32(acc0, a_c, b0_c);
            acc1 = WMMA_F32(acc1, a_c, b1_c);
            acc2 = WMMA_F32(acc2, a_c, b2_c);
            acc3 = WMMA_F32(acc3, a_c, b3_c);

            a_c = a_n;
            b0_c = b0_n; b1_c = b1_n; b2_c = b2_n; b3_c = b3_n;
        }
        acc0 = WMMA_F32(acc0, a_c, b0_c);
        acc1 = WMMA_F32(acc1, a_c, b1_c);
        acc2 = WMMA_F32(acc2, a_c, b2_c);
        acc3 = WMMA_F32(acc3, a_c, b3_c);
    } else {
        // Edge tile: clamped per-tile column pointers (constant row stride).
        int c0 = n0 +  0 + r, c1 = n0 + 16 + r, c2 = n0 + 32 + r, c3 = n0 + 48 + r;
        if (c0 > N - 1) c0 = N - 1;
        if (c1 > N - 1) c1 = N - 1;
        if (c2 > N - 1) c2 = N - 1;
        if (c3 > N - 1) c3 = N - 1;
        const size_t rb = (size_t)(2 * half) * N;
        const float* p0 = B + rb + c0;
        const float* p1 = B + rb + c1;
        const float* p2 = B + rb + c2;
        const float* p3 = B + rb + c3;

        v2f a_c = *(const v2f*)ap;  ap += 4;
        v2f b0_c, b1_c, b2_c, b3_c;
        b0_c.x = p0[0]; b0_c.y = p0[N];
        b1_c.x = p1[0]; b1_c.y = p1[N];
        b2_c.x = p2[0]; b2_c.y = p2[N];
        b3_c.x = p3[0]; b3_c.y = p3[N];
        p0 += 4 * N; p1 += 4 * N; p2 += 4 * N; p3 += 4 * N;

        for (int k = 0; k + 4 < K; k += 4) {
            const v2f a_n = *(const v2f*)ap;  ap += 4;
            v2f b0_n, b1_n, b2_n, b3_n;
            b0_n.x = p0[0]; b0_n.y = p0[N];
            b1_n.x = p1[0]; b1_n.y = p1[N];
            b2_n.x = p2[0]; b2_n.y = p2[N];
            b3_n.x = p3[0]; b3_n.y = p3[N];
            p0 += 4 * N; p1 += 4 * N; p2 += 4 * N; p3 += 4 * N;

            acc0 = WMMA_F32(acc0, a_c, b0_c);
            acc1 = WMMA_F32(acc1, a_c, b1_c);
            acc2 = WMMA_F32(acc2, a_c, b2_c);
            acc3 = WMMA_F32(acc3, a_c, b3_c);

            a_c = a_n;
            b0_c = b0_n; b1_c = b1_n; b2_c = b2_n; b3_c = b3_n;
        }
        acc0 = WMMA_F32(acc0, a_c, b0_c);
        acc1 = WMMA_F32(acc1, a_c, b1_c);
        acc2 = WMMA_F32(acc2, a_c, b2_c);
        acc3 = WMMA_F32(acc3, a_c, b3_c);
    }

    // ---- bias + optional ReLU, guarded stores ----
    v8f accs[4] = { acc0, acc1, acc2, acc3 };
#pragma unroll
    for (int t = 0; t < 4; ++t) {
        const int col = n0 + t * 16 + r;
        if (!EDGE || col < N) {
            const float bv = bias[col];
#pragma unroll
            for (int i = 0; i < 8; ++i) {
                const int row = m0 + half * 8 + i;
                if (row < NROIS) {
                    float v = accs[t][i] + bv;
                    if (RELU) v = fmaxf(v, 0.0f);
                    Cout[(size_t)row * N + col] = v;
                }
            }
        }
    }
}

template <int N, int K, int RELU>
__global__ void roihead_gemm_wmma_kernel(const float* __restrict__ A,
                                         const float* __restrict__ B,
                                         const float* __restrict__ bias,
                                         float* __restrict__ Cout)
{
    const int m0 = (blockIdx.x * 4 + threadIdx.y) * 16;
    const int n0 = blockIdx.y * 64;
    if (n0 + 63 < N) {
        gemm_wmma_body<N, K, RELU, false>(A, B, bias, Cout, m0, n0);
    } else {
        gemm_wmma_body<N, K, RELU, true>(A, B, bias, Cout, m0, n0);
    }
}

// ---------------------------------------------------------------------------
extern "C" void kernel_launch(void* const* d_in, const int* in_sizes, int n_in,
                              void* d_out, int out_size, void* d_ws, size_t ws_size,
                              hipStream_t stream) {
    const float* feat2 = (const float*)d_in[0];
    const float* feat3 = (const float*)d_in[1];
    const float* feat4 = (const float*)d_in[2];
    const float* feat5 = (const float*)d_in[3];
    const float* rois  = (const float*)d_in[4];
    const int*   imgsz = (const int*)  d_in[5];
    const float* W1    = (const float*)d_in[6];
    const float* b1    = (const float*)d_in[7];
    const float* W2    = (const float*)d_in[8];
    const float* b2    = (const float*)d_in[9];
    const float* Wloc  = (const float*)d_in[10];
    const float* bloc  = (const float*)d_in[11];
    const float* Wsc   = (const float*)d_in[12];
    const float* bsc   = (const float*)d_in[13];

    float* out = (float*)d_out;            // [1000*324] locs, then [1000*81] scores

    float* xbuf = (float*)d_ws;                        // 1000 x 12544
    float* h1   = xbuf + (size_t)NROIS * RD;           // 1000 x 1024
    float* h2   = h1 + (size_t)NROIS * 1024;           // 1000 x 1024

    const dim3 gblock(32, 4);
    const int  gx = (NROIS + 63) / 64;

    // 1) ROI align -> x
    roihead_roialign_kernel<<<NROIS, 256, 0, stream>>>(feat2, feat3, feat4, feat5,
                                                       rois, imgsz, xbuf);

    // 2) fc1: x @ W1 + b1, ReLU   (1000 x 12544 x 1024)
    roihead_gemm_wmma_kernel<1024, RD, 1>
        <<<dim3(gx, 1024 / 64), gblock, 0, stream>>>(xbuf, W1, b1, h1);

    // 3) fc2: h1 @ W2 + b2, ReLU  (1000 x 1024 x 1024)
    roihead_gemm_wmma_kernel<1024, 1024, 1>
        <<<dim3(gx, 1024 / 64), gblock, 0, stream>>>(h1, W2, b2, h2);

    // 4) loc head: h2 @ Wloc + bloc   (1000 x 1024 x 324)
    roihead_gemm_wmma_kernel<324, 1024, 0>
        <<<dim3(gx, (324 + 63) / 64), gblock, 0, stream>>>(h2, Wloc, bloc, out);

    // 5) score head: h2 @ Wsc + bsc   (1000 x 1024 x 81)
    roihead_gemm_wmma_kernel<81, 1024, 0>
        <<<dim3(gx, (81 + 63) / 64), gblock, 0, stream>>>(h2, Wsc, bsc,
                                                          out + (size_t)NROIS * 324);
}